// SymmetricTransitionDownBlockPaperv3_9242769621757
// MI455X (gfx1250) — compile-verified
//
#include <hip/hip_runtime.h>
#include <hip/hip_bf16.h>

// ---------------- problem constants ----------------
constexpr int kN     = 16384;
constexpr int kCIN   = 64;
constexpr int kCOUT  = 128;
constexpr int kK     = 16;      // neighbors
constexpr int kM     = 4096;    // N / stride
constexpr int kR     = kM * kK; // 65536 rows of x_knn
constexpr int kKP    = 68;      // 3+64=67 padded to multiple of 4 for f32 WMMA
constexpr float kEPS = 1e-5f;

// output layout (floats): n_p [M*3] | y_out [M*128] | n_o [1]
constexpr int OUT_NP   = 0;
constexpr int OUT_Y    = kM * 3;              // 12288
constexpr int OUT_NO   = OUT_Y + kM * kCOUT;  // 536576

// ---------------- workspace layout (bytes, 256-aligned) ----------------
constexpr size_t OFF_IDX  = 0;          // int[4096]
constexpr size_t OFF_KNN  = 16384;      // int[65536]
constexpr size_t OFF_WL2T = 278528;     // f32 [128][64]   (w_lin2 transposed)
constexpr size_t OFF_WS1T = 311296;     // f32 [64][68]    (w_shr1 transposed+padded)
constexpr size_t OFF_XK   = 328704;     // f32 [65536][68]
constexpr size_t OFF_Y    = 18154496;   // f32 [16384][128]
constexpr size_t OFF_H    = 26543104;   // f32 [65536][64]
constexpr size_t OFF_S    = 43320320;   // f32 [65536]
constexpr size_t OFF_WSM  = 43582464;   // f32 [65536]
constexpr size_t OFF_STY  = 43844608;   // f32 [256]  (mean | rstd) for y
constexpr size_t OFF_STH  = 43845632;   // f32 [128]  (mean | rstd) for h

typedef __attribute__((ext_vector_type(2))) float v2f;
typedef __attribute__((ext_vector_type(8))) float v8f;

// =====================================================================
// 1) Furthest point sampling: single persistent block, coords in regs.
// =====================================================================
__global__ __launch_bounds__(1024) void fps_kernel(const float* __restrict__ p,
                                                   int* __restrict__ idx,
                                                   float* __restrict__ np_out) {
  const int tid  = threadIdx.x;
  const int lane = tid & 31;
  const int wv   = tid >> 5;
  __shared__ float s_last[3];
  __shared__ float s_bv[32];
  __shared__ int   s_bi[32];

  float px[16], py[16], pz[16], dmin[16];
#pragma unroll
  for (int i = 0; i < 16; ++i) {
    int j = i * 1024 + tid;
    px[i] = p[j * 3 + 0]; py[i] = p[j * 3 + 1]; pz[i] = p[j * 3 + 2];
    dmin[i] = 1e10f;
  }
  if (tid == 0) {
    idx[0] = 0;
    s_last[0] = p[0]; s_last[1] = p[1]; s_last[2] = p[2];
  }
  __syncthreads();

  for (int t = 1; t < kM; ++t) {
    const float lx = s_last[0], ly = s_last[1], lz = s_last[2];
    float bv = -1.0f; int bi = 0;
#pragma unroll
    for (int i = 0; i < 16; ++i) {
      float dx = px[i] - lx, dy = py[i] - ly, dz = pz[i] - lz;
      float d = dx * dx + dy * dy + dz * dz;
      dmin[i] = fminf(dmin[i], d);
      if (dmin[i] > bv) { bv = dmin[i]; bi = i * 1024 + tid; }
    }
    // wave32 argmax (ties -> lower index)
#pragma unroll
    for (int off = 16; off > 0; off >>= 1) {
      float ov = __shfl_down(bv, off, 32);
      int   oi = __shfl_down(bi, off, 32);
      if (ov > bv || (ov == bv && oi < bi)) { bv = ov; bi = oi; }
    }
    if (lane == 0) { s_bv[wv] = bv; s_bi[wv] = bi; }
    __syncthreads();
    if (wv == 0) {
      bv = s_bv[lane]; bi = s_bi[lane];
#pragma unroll
      for (int off = 16; off > 0; off >>= 1) {
        float ov = __shfl_down(bv, off, 32);
        int   oi = __shfl_down(bi, off, 32);
        if (ov > bv || (ov == bv && oi < bi)) { bv = ov; bi = oi; }
      }
      if (lane == 0) {
        idx[t] = bi;
        s_last[0] = p[bi * 3 + 0];
        s_last[1] = p[bi * 3 + 1];
        s_last[2] = p[bi * 3 + 2];
      }
    }
    __syncthreads();
  }
  // gather n_p into the output
  for (int m = tid; m < kM; m += 1024) {
    int j = idx[m];
    np_out[m * 3 + 0] = p[j * 3 + 0];
    np_out[m * 3 + 1] = p[j * 3 + 1];
    np_out[m * 3 + 2] = p[j * 3 + 2];
  }
}

// =====================================================================
// 2) KNN: one thread per center; LDS-tiled scan + register top-16.
// =====================================================================
__global__ __launch_bounds__(256) void knn_kernel(const float* __restrict__ p,
                                                  const float* __restrict__ np,
                                                  int* __restrict__ knn) {
  const int m = blockIdx.x * 256 + threadIdx.x; // 0..4095
  const float cx = np[m * 3 + 0], cy = np[m * 3 + 1], cz = np[m * 3 + 2];
  __shared__ float sp[512 * 3];

  float bd[kK]; int bix[kK];
#pragma unroll
  for (int q = 0; q < kK; ++q) { bd[q] = 3.0e38f; bix[q] = 0; }

  for (int tile = 0; tile < kN; tile += 512) {
    for (int i = threadIdx.x; i < 512 * 3; i += 256) sp[i] = p[tile * 3 + i];
    __syncthreads();
    for (int j = 0; j < 512; ++j) {
      float dx = sp[j * 3 + 0] - cx;
      float dy = sp[j * 3 + 1] - cy;
      float dz = sp[j * 3 + 2] - cz;
      float d = dx * dx + dy * dy + dz * dz;
      if (d < bd[kK - 1]) {
        bd[kK - 1] = d; bix[kK - 1] = tile + j;
#pragma unroll
        for (int q = kK - 1; q > 0; --q) {
          if (bd[q] < bd[q - 1]) {
            float tv = bd[q];  bd[q]  = bd[q - 1];  bd[q - 1]  = tv;
            int   ti = bix[q]; bix[q] = bix[q - 1]; bix[q - 1] = ti;
          }
        }
      }
    }
    __syncthreads();
  }
#pragma unroll
  for (int q = 0; q < kK; ++q) knn[m * kK + q] = bix[q];
}

// =====================================================================
// 3) Weight prep: transpose to [N][K] f32 (pad K: 67 -> 68).
// =====================================================================
__global__ void prep_weights(const float* __restrict__ w_lin2,
                             const float* __restrict__ w_shr1,
                             float* __restrict__ wl2t,
                             float* __restrict__ ws1t) {
  // w_lin2 [64][128] -> wl2t [128][64]
  for (int t = threadIdx.x; t < kCOUT * kCIN; t += blockDim.x) {
    int n = t / kCIN, k = t % kCIN;
    wl2t[t] = w_lin2[k * kCOUT + n];
  }
  // w_shr1 [67][64] -> ws1t [64][68] (zero-padded)
  for (int t = threadIdx.x; t < kCIN * kKP; t += blockDim.x) {
    int n = t / kKP, k = t % kKP;
    ws1t[t] = (k < 67) ? w_shr1[k * kCIN + n] : 0.0f;
  }
}

// =====================================================================
// 4) Build x_knn rows: [rel_xyz | x[j] | pad] f32, 68 cols.
// =====================================================================
__global__ __launch_bounds__(128) void build_xk(const float* __restrict__ p,
                                                const float* __restrict__ x,
                                                const float* __restrict__ np,
                                                const int* __restrict__ knn,
                                                float* __restrict__ xk) {
  const int r = blockIdx.x * 128 + threadIdx.x; // 0..65535
  const int m = r >> 4;
  const int j = knn[r];
  float* row = xk + (size_t)r * kKP;
  row[0] = p[j * 3 + 0] - np[m * 3 + 0];
  row[1] = p[j * 3 + 1] - np[m * 3 + 1];
  row[2] = p[j * 3 + 2] - np[m * 3 + 2];
#pragma unroll 8
  for (int c = 0; c < kCIN; ++c) row[3 + c] = x[j * kCIN + c];
  row[67] = 0.0f;
}

// =====================================================================
// 5) WMMA GEMM (full fp32): C[Mr,Nc] = A[Mr,Kc] * Bt[Nc,Kc]^T.
//    One wave per 16x16 C tile; v_wmma_f32_16x16x4_f32 per K-step.
//    A frag (16x4): lane<16 -> (m=lane, k=kb+0,1); lane>=16 -> k=kb+2,3
//    B frag (4x16): lane<16 -> (n=lane, k=kb+0,1); lane>=16 -> k=kb+2,3
//    -> both are contiguous float2 at [row*Kc + kb + 2*hi].
// =====================================================================
__global__ __launch_bounds__(256) void gemm_wmma_f32(const float* __restrict__ A,
                                                     const float* __restrict__ Bt,
                                                     float* __restrict__ C,
                                                     int Mr, int Nc, int Kc) {
  const int lane = threadIdx.x & 31;
  const int wave = threadIdx.x >> 5;
  const int tilesN = Nc >> 4;
  const int totalTiles = (Mr >> 4) * tilesN;
  const int tile = blockIdx.x * (blockDim.x >> 5) + wave;
  if (tile >= totalTiles) return;   // wave-uniform: EXEC stays all-ones

  const int tm  = (tile / tilesN) << 4;
  const int tn  = (tile % tilesN) << 4;
  const int hi  = lane >> 4;        // 0: low half-lanes, 1: high
  const int l15 = lane & 15;

  const float* arow = A  + (size_t)(tm + l15) * Kc + 2 * hi;
  const float* brow = Bt + (size_t)(tn + l15) * Kc + 2 * hi;

  v8f acc = {};
  for (int kb = 0; kb < Kc; kb += 4) {
    v2f a = *reinterpret_cast<const v2f*>(arow + kb);
    v2f b = *reinterpret_cast<const v2f*>(brow + kb);
    acc = __builtin_amdgcn_wmma_f32_16x16x4_f32(
        /*neg_a=*/false, a, /*neg_b=*/false, b,
        /*c_mod=*/(short)0, acc, /*reuse_a=*/false, /*reuse_b=*/false);
  }
  // C/D layout: vgpr r -> row tm + r + 8*hi, col tn + (lane&15)
#pragma unroll
  for (int r = 0; r < 8; ++r)
    C[(size_t)(tm + r + 8 * hi) * Nc + tn + l15] = acc[r];
}

// =====================================================================
// 6) BatchNorm stats: one block per column (deterministic, no atomics).
//    stats[col] = mean, stats[Ncols+col] = rsqrt(var+eps)
// =====================================================================
__global__ __launch_bounds__(256) void bn_stats(const float* __restrict__ Cm,
                                                int R, int Ncols,
                                                float* __restrict__ stats) {
  const int col = blockIdx.x;
  float s = 0.0f, s2 = 0.0f;
  for (int r = threadIdx.x; r < R; r += 256) {
    float v = Cm[(size_t)r * Ncols + col];
    s += v; s2 += v * v;
  }
  __shared__ float rs[256], rq[256];
  rs[threadIdx.x] = s; rq[threadIdx.x] = s2;
  __syncthreads();
  for (int off = 128; off > 0; off >>= 1) {
    if (threadIdx.x < off) {
      rs[threadIdx.x] += rs[threadIdx.x + off];
      rq[threadIdx.x] += rq[threadIdx.x + off];
    }
    __syncthreads();
  }
  if (threadIdx.x == 0) {
    float mu  = rs[0] / (float)R;
    float var = rq[0] / (float)R - mu * mu;
    stats[col]         = mu;
    stats[Ncols + col] = rsqrtf(var + kEPS);
  }
}

__global__ void bn_relu_apply(float* __restrict__ Cm, int R, int Ncols,
                              const float* __restrict__ stats,
                              const float* __restrict__ g,
                              const float* __restrict__ b) {
  const int total = R * Ncols;
  for (int i = blockIdx.x * blockDim.x + threadIdx.x; i < total;
       i += gridDim.x * blockDim.x) {
    int col = i % Ncols;
    float v = (Cm[i] - stats[col]) * stats[Ncols + col] * g[col] + b[col];
    Cm[i] = fmaxf(v, 0.0f);
  }
}

// =====================================================================
// 7) shrinker head: s[r] = h[r,:] . w_shr2 + b ; softmax over K per m.
// =====================================================================
__global__ __launch_bounds__(256) void shr2_kernel(const float* __restrict__ h,
                                                   const float* __restrict__ w2,
                                                   const float* __restrict__ b2,
                                                   float* __restrict__ s) {
  const int r = blockIdx.x * 256 + threadIdx.x;
  const float* row = h + (size_t)r * kCIN;
  float acc = b2[0];
#pragma unroll 8
  for (int c = 0; c < kCIN; ++c) acc += row[c] * w2[c];
  s[r] = acc;
}

__global__ __launch_bounds__(256) void softmax_kernel(const float* __restrict__ s,
                                                      float* __restrict__ w) {
  const int m = blockIdx.x * 256 + threadIdx.x; // 0..4095
  float mx = -3.0e38f;
#pragma unroll
  for (int q = 0; q < kK; ++q) mx = fmaxf(mx, s[m * kK + q]);
  float e[kK]; float sum = 0.0f;
#pragma unroll
  for (int q = 0; q < kK; ++q) { e[q] = __expf(s[m * kK + q] - mx); sum += e[q]; }
  float inv = 1.0f / sum;
#pragma unroll
  for (int q = 0; q < kK; ++q) w[m * kK + q] = e[q] * inv;
}

// =====================================================================
// 8) Output: y_out[m,c] = sum_k y[knn[m,k], c] * w[m,k] ; write n_o.
// =====================================================================
__global__ __launch_bounds__(128) void out_kernel(const float* __restrict__ y,
                                                  const int* __restrict__ knn,
                                                  const float* __restrict__ wsm,
                                                  float* __restrict__ out) {
  const int m = blockIdx.x;
  const int c = threadIdx.x;
  float acc = 0.0f;
#pragma unroll
  for (int q = 0; q < kK; ++q) {
    int j = knn[m * kK + q];
    acc += y[(size_t)j * kCOUT + c] * wsm[m * kK + q];
  }
  out[OUT_Y + m * kCOUT + c] = acc;
  if (m == 0 && c == 0) out[OUT_NO] = (float)kM;
}

// =====================================================================
extern "C" void kernel_launch(void* const* d_in, const int* in_sizes, int n_in,
                              void* d_out, int out_size, void* d_ws, size_t ws_size,
                              hipStream_t stream) {
  const float* p       = (const float*)d_in[0];
  const float* x       = (const float*)d_in[1];
  // d_in[2] = o (unused: single batch)
  const float* w_lin2  = (const float*)d_in[3];
  const float* g2      = (const float*)d_in[4];
  const float* b2      = (const float*)d_in[5];
  const float* w_shr1  = (const float*)d_in[6];
  const float* g_s     = (const float*)d_in[7];
  const float* b_s     = (const float*)d_in[8];
  const float* w_shr2  = (const float*)d_in[9];
  const float* b_shr2  = (const float*)d_in[10];
  float* out = (float*)d_out;

  char* ws = (char*)d_ws;
  int*    idx  = (int*)(ws + OFF_IDX);
  int*    knn  = (int*)(ws + OFF_KNN);
  float*  wl2t = (float*)(ws + OFF_WL2T);
  float*  ws1t = (float*)(ws + OFF_WS1T);
  float*  xk   = (float*)(ws + OFF_XK);
  float*  yb   = (float*)(ws + OFF_Y);
  float*  hb   = (float*)(ws + OFF_H);
  float*  sb   = (float*)(ws + OFF_S);
  float*  wsm  = (float*)(ws + OFF_WSM);
  float*  sty  = (float*)(ws + OFF_STY);
  float*  sth  = (float*)(ws + OFF_STH);

  float* np_out = out + OUT_NP;

  // independent prep
  prep_weights<<<1, 256, 0, stream>>>(w_lin2, w_shr1, wl2t, ws1t);

  // FPS -> n_p, idx
  fps_kernel<<<1, 1024, 0, stream>>>(p, idx, np_out);

  // KNN
  knn_kernel<<<kM / 256, 256, 0, stream>>>(p, np_out, knn);

  // y = relu(BN(x @ w_lin2))  -- fp32 WMMA GEMM 16384x128x64 (A = x directly)
  {
    const int tiles = (kN / 16) * (kCOUT / 16); // 8192
    gemm_wmma_f32<<<tiles / 8, 256, 0, stream>>>(x, wl2t, yb, kN, kCOUT, kCIN);
    bn_stats<<<kCOUT, 256, 0, stream>>>(yb, kN, kCOUT, sty);
    bn_relu_apply<<<2048, 256, 0, stream>>>(yb, kN, kCOUT, sty, g2, b2);
  }

  // x_knn gather + h = relu(BN(x_knn @ w_shr1)) -- fp32 WMMA GEMM 65536x64x68
  build_xk<<<kR / 128, 128, 0, stream>>>(p, x, np_out, knn, xk);
  {
    const int tiles = (kR / 16) * (kCIN / 16); // 16384
    gemm_wmma_f32<<<tiles / 8, 256, 0, stream>>>(xk, ws1t, hb, kR, kCIN, kKP);
    bn_stats<<<kCIN, 256, 0, stream>>>(hb, kR, kCIN, sth);
    bn_relu_apply<<<4096, 256, 0, stream>>>(hb, kR, kCIN, sth, g_s, b_s);
  }

  // s = h @ w_shr2 + b ; softmax over neighbors
  shr2_kernel<<<kR / 256, 256, 0, stream>>>(hb, w_shr2, b_shr2, sb);
  softmax_kernel<<<kM / 256, 256, 0, stream>>>(sb, wsm);

  // y_out = sum_k y[knn] * w ; n_o
  out_kernel<<<kM, 128, 0, stream>>>(yb, knn, wsm, out);
}